// GAT_Net_ASAP_72060961292410
// MI455X (gfx1250) — compile-verified
//
#include <hip/hip_runtime.h>
#include <hip/hip_bf16.h>

typedef __attribute__((ext_vector_type(2))) float v2f;
typedef __attribute__((ext_vector_type(8))) float v8f;

// ---- problem constants (fixed by reference) ----
#define NND   32768      // N nodes
#define CCH   128        // C channels
#define NHEAD 2
#define DHH   64
#define NB    32         // graphs
#define NPG   1024       // nodes per graph
#define KSEL  512        // k = NP/2
#define E0C   262144     // edges w/o self loops
#define ETC   294912     // edges + self loops

__device__ __forceinline__ float lrelu02(float v) { return v >= 0.0f ? v : 0.2f * v; }

__device__ __forceinline__ void atomicMaxF(float* addr, float v) {
    // IEEE monotonic trick: ints compare like floats for >=0, reversed for <0
    if (v >= 0.0f) atomicMax((int*)addr, __float_as_int(v));
    else           atomicMin((unsigned int*)addr, __float_as_uint(v));
}

__device__ __forceinline__ void edge_sd(const int* __restrict__ ei, int e, int& s, int& d) {
    if (e < E0C) { s = ei[e]; d = ei[E0C + e]; }
    else         { s = e - E0C; d = e - E0C; }
}

// =============== generic fill ===============
__global__ void fill_u32_kernel(unsigned int* __restrict__ p, unsigned int bits, long long n) {
    long long i = (long long)blockIdx.x * blockDim.x + threadIdx.x;
    long long stride = (long long)gridDim.x * blockDim.x;
    for (; i < n; i += stride) p[i] = bits;
}

// =============== fp32 WMMA GEMM: C = A(MxK) @ B(KxN), row-major ===============
// one wave per 16x16 tile, V_WMMA_F32_16X16X4_F32, batch via blockIdx.z
__global__ void wmma_gemm_f32_kernel(const float* __restrict__ A, const float* __restrict__ B,
                                     float* __restrict__ C,
                                     int Mdim, int Ndim, int Kdim,
                                     int lda, int ldb, int ldc,
                                     long long sA, long long sB, long long sC) {
    A += (long long)blockIdx.z * sA;
    B += (long long)blockIdx.z * sB;
    C += (long long)blockIdx.z * sC;
    int wave = (blockIdx.x * blockDim.x + threadIdx.x) >> 5;
    int lane = threadIdx.x & 31;
    int tilesN = Ndim >> 4;
    int nTiles = (Mdim >> 4) * tilesN;
    if (wave >= nTiles) return;                 // whole-wave exit: EXEC stays all-ones
    int tm = wave / tilesN, tn = wave - tm * tilesN;
    int row0 = tm << 4, col0 = tn << 4;
    int half = lane >> 4, l16 = lane & 15;

    v8f acc = {0.f, 0.f, 0.f, 0.f, 0.f, 0.f, 0.f, 0.f};
    const float* arow = A + (long long)(row0 + l16) * lda + (half << 1);
    const float* bcol = B + (long long)(half << 1) * ldb + col0 + l16;
    for (int k = 0; k < Kdim; k += 4) {
        v2f a = *reinterpret_cast<const v2f*>(arow);   // A[r, k+2*half .. +1], 8B aligned
        v2f b;
        b.x = bcol[0];                                  // B[k+2*half,   col]
        b.y = bcol[ldb];                                // B[k+2*half+1, col]
        acc = __builtin_amdgcn_wmma_f32_16x16x4_f32(false, a, false, b,
                                                    (short)0, acc, false, false);
        arow += 4;
        bcol += 4 * ldb;
    }
    float* crow = C + (long long)(row0 + half * 8) * ldc + col0 + l16;
#pragma unroll
    for (int r = 0; r < 8; ++r) crow[(long long)r * ldc] = acc[r];
}

// =============== GAT1 per-node head coefficients ===============
__global__ void node_head_dots(const float* __restrict__ h, const float* __restrict__ att_src,
                               const float* __restrict__ att_dst,
                               float* __restrict__ as_, float* __restrict__ ad_) {
    int t = blockIdx.x * blockDim.x + threadIdx.x;
    if (t >= NND * NHEAD) return;
    int n = t >> 1, hh = t & 1;
    const float* row = h + (long long)n * CCH + hh * DHH;
    const float* ws = att_src + hh * DHH;
    const float* wd = att_dst + hh * DHH;
    float ss = 0.f, dd = 0.f;
    for (int c = 0; c < DHH; ++c) { float v = row[c]; ss += v * ws[c]; dd += v * wd[c]; }
    as_[t] = ss; ad_[t] = dd;
}

__global__ void edge_max1(const int* __restrict__ ei, const float* __restrict__ as_,
                          const float* __restrict__ ad_, float* __restrict__ mx1) {
    int e = blockIdx.x * blockDim.x + threadIdx.x;
    if (e >= ETC) return;
    int s, d; edge_sd(ei, e, s, d);
#pragma unroll
    for (int hh = 0; hh < NHEAD; ++hh) {
        float v = lrelu02(as_[s * NHEAD + hh] + ad_[d * NHEAD + hh]);
        atomicMaxF(&mx1[d * NHEAD + hh], v);
    }
}

__global__ void edge_exp1(const int* __restrict__ ei, const float* __restrict__ as_,
                          const float* __restrict__ ad_, const float* __restrict__ mx1,
                          float* __restrict__ ex1, float* __restrict__ den1) {
    int e = blockIdx.x * blockDim.x + threadIdx.x;
    if (e >= ETC) return;
    int s, d; edge_sd(ei, e, s, d);
#pragma unroll
    for (int hh = 0; hh < NHEAD; ++hh) {
        float v = lrelu02(as_[s * NHEAD + hh] + ad_[d * NHEAD + hh]);
        float ex = __expf(v - mx1[d * NHEAD + hh]);
        ex1[e * NHEAD + hh] = ex;
        atomicAdd(&den1[d * NHEAD + hh], ex);
    }
}

// one block per edge, 128 threads = channels
__global__ void x1_accum(const int* __restrict__ ei, const float* __restrict__ h,
                         const float* __restrict__ ex1, const float* __restrict__ den1,
                         float* __restrict__ x1) {
    int e = blockIdx.x;
    int c = threadIdx.x;
    int s, d; edge_sd(ei, e, s, d);
    int hh = c >> 6;
    float alpha = ex1[e * NHEAD + hh] / den1[d * NHEAD + hh];
    atomicAdd(&x1[(long long)d * CCH + c], alpha * h[(long long)s * CCH + c]);
}

__global__ void relu_bias(float* __restrict__ x1, const float* __restrict__ bias1) {
    long long t = (long long)blockIdx.x * blockDim.x + threadIdx.x;
    if (t >= (long long)NND * CCH) return;
    float v = x1[t] + bias1[t & (CCH - 1)];
    x1[t] = v > 0.f ? v : 0.f;
}

// one block per edge, 128 threads: segment max of x1[src] into xmax[dst]
__global__ void xmax_accum(const int* __restrict__ ei, const float* __restrict__ x1,
                           float* __restrict__ xmax) {
    int e = blockIdx.x;
    int c = threadIdx.x;
    int s, d; edge_sd(ei, e, s, d);
    atomicMaxF(&xmax[(long long)d * CCH + c], x1[(long long)s * CCH + c]);
}

// q1[n] = dot(xq[n]+b_lin, W_att[0:C]) ; q2[n] = dot(x1[n], W_att[C:2C])
__global__ void q1q2_kernel(const float* __restrict__ xq, const float* __restrict__ x1,
                            const float* __restrict__ b_lin, const float* __restrict__ W_att,
                            float* __restrict__ q1, float* __restrict__ q2) {
    int n = blockIdx.x * blockDim.x + threadIdx.x;
    if (n >= NND) return;
    const float* r1 = xq + (long long)n * CCH;
    const float* r2 = x1 + (long long)n * CCH;
    float a = 0.f, b = 0.f;
    for (int c = 0; c < CCH; ++c) {
        a += (r1[c] + b_lin[c]) * W_att[c];
        b += r2[c] * W_att[CCH + c];
    }
    q1[n] = a; q2[n] = b;
}

__global__ void edge_max2(const int* __restrict__ ei, const float* __restrict__ q1,
                          const float* __restrict__ q2, const float* __restrict__ b_att,
                          float* __restrict__ mx2) {
    int e = blockIdx.x * blockDim.x + threadIdx.x;
    if (e >= ETC) return;
    int s, d; edge_sd(ei, e, s, d);
    atomicMaxF(&mx2[d], lrelu02(q1[d] + q2[s] + b_att[0]));
}

__global__ void edge_exp2(const int* __restrict__ ei, const float* __restrict__ q1,
                          const float* __restrict__ q2, const float* __restrict__ b_att,
                          const float* __restrict__ mx2, float* __restrict__ sex,
                          float* __restrict__ den2) {
    int e = blockIdx.x * blockDim.x + threadIdx.x;
    if (e >= ETC) return;
    int s, d; edge_sd(ei, e, s, d);
    float ex = __expf(lrelu02(q1[d] + q2[s] + b_att[0]) - mx2[d]);
    sex[e] = ex;
    atomicAdd(&den2[d], ex);
}

__global__ void score_scale(const int* __restrict__ ei, const float* __restrict__ den2,
                            float* __restrict__ sex) {
    int e = blockIdx.x * blockDim.x + threadIdx.x;
    if (e >= ETC) return;
    int s, d; edge_sd(ei, e, s, d);
    sex[e] = sex[e] / den2[d];
}

// one block per edge, 128 threads: xc[dst] += score * x1[src]
__global__ void xc_accum(const int* __restrict__ ei, const float* __restrict__ x1,
                         const float* __restrict__ score, float* __restrict__ xc) {
    int e = blockIdx.x;
    int c = threadIdx.x;
    int s, d; edge_sd(ei, e, s, d);
    atomicAdd(&xc[(long long)d * CCH + c], score[e] * x1[(long long)s * CCH + c]);
}

__global__ void le_dots(const float* __restrict__ xc, const float* __restrict__ W1v,
                        const float* __restrict__ b1v, const float* __restrict__ W2v,
                        const float* __restrict__ W3v, const float* __restrict__ b3v,
                        float* __restrict__ a_n, float* __restrict__ bb_n,
                        float* __restrict__ cle3) {
    int n = blockIdx.x * blockDim.x + threadIdx.x;
    if (n >= NND) return;
    const float* row = xc + (long long)n * CCH;
    float a = 0.f, b = 0.f, c3 = 0.f;
    for (int c = 0; c < CCH; ++c) { float v = row[c]; a += v * W1v[c]; b += v * W2v[c]; c3 += v * W3v[c]; }
    a_n[n] = a + b1v[0]; bb_n[n] = b; cle3[n] = c3 + b3v[0];
}

__global__ void agg_accum(const int* __restrict__ ei, const float* __restrict__ a_n,
                          const float* __restrict__ bb_n, float* __restrict__ agg) {
    int e = blockIdx.x * blockDim.x + threadIdx.x;
    if (e >= ETC) return;
    int s, d; edge_sd(ei, e, s, d);
    atomicAdd(&agg[d], a_n[s] - bb_n[d]);
}

__global__ void fitness_kernel(const float* __restrict__ agg, const float* __restrict__ cle3,
                               float* __restrict__ fit) {
    int n = blockIdx.x * blockDim.x + threadIdx.x;
    if (n >= NND) return;
    float z = agg[n] + cle3[n];
    fit[n] = 1.0f / (1.0f + __expf(-z));
}

// per-graph exact top-k selection by rank counting (ties broken by lower index),
// compacted in ascending node order (final output is permutation-invariant)
__global__ void topk_kernel(const float* __restrict__ fit, int* __restrict__ topi,
                            int* __restrict__ selcol) {
    __shared__ float f[NPG];
    __shared__ int sel[NPG];
    int b = blockIdx.x;
    for (int i = threadIdx.x; i < NPG; i += blockDim.x) f[i] = fit[b * NPG + i];
    __syncthreads();
    for (int i = threadIdx.x; i < NPG; i += blockDim.x) {
        float fi = f[i];
        int rank = 0;
        for (int j = 0; j < NPG; ++j) {
            float fj = f[j];
            rank += (fj > fi) || (fj == fi && j < i);
        }
        sel[i] = (rank < KSEL) ? 1 : 0;
    }
    __syncthreads();
    for (int i = threadIdx.x; i < NPG; i += blockDim.x) {
        if (sel[i]) {
            int pos = 0;
            for (int j = 0; j < i; ++j) pos += sel[j];
            topi[b * KSEL + pos] = i;
            selcol[b * NPG + i] = pos;
        } else {
            selcol[b * NPG + i] = -1;
        }
    }
}

__global__ void xsel_build(const float* __restrict__ xc, const float* __restrict__ fit,
                           const int* __restrict__ topi, float* __restrict__ Xsel) {
    long long t = (long long)blockIdx.x * blockDim.x + threadIdx.x;
    if (t >= (long long)NB * KSEL * CCH) return;
    int c = (int)(t & (CCH - 1));
    int kk = (int)((t >> 7) & (KSEL - 1));
    int b = (int)(t >> 16);
    int g = b * NPG + topi[b * KSEL + kk];
    Xsel[t] = xc[(long long)g * CCH + c] * fit[g];
}

// Ssel[b, sl, col] += score  (col = compact column of dst, if selected)
__global__ void ssel_scatter(const int* __restrict__ ei, const float* __restrict__ score,
                             const int* __restrict__ selcol, float* __restrict__ Ssel) {
    int e = blockIdx.x * blockDim.x + threadIdx.x;
    if (e >= ETC) return;
    int s, d; edge_sd(ei, e, s, d);
    int col = selcol[d];
    if (col < 0) return;
    int b = s >> 10, sl = s & (NPG - 1);
    atomicAdd(&Ssel[((long long)(b * NPG + sl)) * KSEL + col], score[e]);
}

// M[b, kk, dl] += Ssel[b, sl, kk]  (Ssel rows are ~9-sparse -> skip zeros)
__global__ void m_scatter(const int* __restrict__ ei, const float* __restrict__ Ssel,
                          float* __restrict__ M) {
    int e = blockIdx.x;
    int kk = threadIdx.x;               // 512 threads
    int s, d; edge_sd(ei, e, s, d);
    int b = s >> 10, sl = s & (NPG - 1), dl = d & (NPG - 1);
    float v = Ssel[((long long)(b * NPG + sl)) * KSEL + kk];
    if (v != 0.0f)
        atomicAdd(&M[((long long)(b * KSEL + kk)) * NPG + dl], v);
}

__global__ void h2_dots(const float* __restrict__ h2, const float* __restrict__ att_dst2,
                        const float* __restrict__ att_src2, float* __restrict__ ds2,
                        float* __restrict__ ss2) {
    int t = blockIdx.x * blockDim.x + threadIdx.x;
    if (t >= NB * KSEL) return;
    const float* row = h2 + (long long)t * CCH;
    float a = 0.f, b = 0.f;
    for (int c = 0; c < CCH; ++c) { float v = row[c]; a += v * att_dst2[c]; b += v * att_src2[c]; }
    ds2[t] = a; ss2[t] = b;
}

// per (graph,row): masked row softmax of GAT2 logits; accumulate column means
// mask[kk,mm] = (Ac[mm,kk] > 0) | (kk==mm)
__global__ void gat2_softmax_cm(const float* __restrict__ Ac, const float* __restrict__ ds2,
                                const float* __restrict__ ss2, float* __restrict__ cm) {
    __shared__ float red[128];
    int b = blockIdx.x >> 9;
    int kk = blockIdx.x & (KSEL - 1);
    const float* AcB = Ac + (long long)b * KSEL * KSEL;
    float dval = ds2[b * KSEL + kk];
    float vals[4];
    float mx = -3.4e38f;
#pragma unroll
    for (int t = 0; t < 4; ++t) {
        int mm = threadIdx.x + t * 128;
        bool m = (mm == kk) || (AcB[(long long)mm * KSEL + kk] > 0.0f);
        float lg = lrelu02(dval + ss2[b * KSEL + mm]);
        float v = m ? lg : -1e9f;
        vals[t] = v;
        mx = fmaxf(mx, v);
    }
    red[threadIdx.x] = mx; __syncthreads();
    for (int s2 = 64; s2 > 0; s2 >>= 1) {
        if (threadIdx.x < s2) red[threadIdx.x] = fmaxf(red[threadIdx.x], red[threadIdx.x + s2]);
        __syncthreads();
    }
    mx = red[0]; __syncthreads();
    float sum = 0.f;
#pragma unroll
    for (int t = 0; t < 4; ++t) { vals[t] = __expf(vals[t] - mx); sum += vals[t]; }
    red[threadIdx.x] = sum; __syncthreads();
    for (int s2 = 64; s2 > 0; s2 >>= 1) {
        if (threadIdx.x < s2) red[threadIdx.x] += red[threadIdx.x + s2];
        __syncthreads();
    }
    float inv = 1.0f / (red[0] * (float)KSEL);   // fold mean over kk into the weight
#pragma unroll
    for (int t = 0; t < 4; ++t) {
        int mm = threadIdx.x + t * 128;
        atomicAdd(&cm[b * KSEL + mm], vals[t] * inv);
    }
}

// out[b,c] = sum_mm cm[b,mm] * h2[b,mm,c] + bias2[c]
__global__ void final_out(const float* __restrict__ cm, const float* __restrict__ h2,
                          const float* __restrict__ bias2, float* __restrict__ out) {
    int b = blockIdx.x;
    int c = threadIdx.x;
    float acc = bias2[c];
    const float* hb = h2 + (long long)b * KSEL * CCH;
    for (int mm = 0; mm < KSEL; ++mm)
        acc += cm[b * KSEL + mm] * hb[(long long)mm * CCH + c];
    out[b * CCH + c] = acc;
}

// ==================== host ====================
static void launch_gemm(const float* A, const float* B, float* C,
                        int M, int N, int K, int lda, int ldb, int ldc,
                        long long sA, long long sB, long long sC, int batch,
                        hipStream_t stream) {
    int tiles = (M >> 4) * (N >> 4);
    int blocks = (tiles + 7) / 8;   // 8 waves (wave32) per 256-thread block
    wmma_gemm_f32_kernel<<<dim3(blocks, 1, batch), 256, 0, stream>>>(
        A, B, C, M, N, K, lda, ldb, ldc, sA, sB, sC);
}

extern "C" void kernel_launch(void* const* d_in, const int* in_sizes, int n_in,
                              void* d_out, int out_size, void* d_ws, size_t ws_size,
                              hipStream_t stream) {
    const float* x        = (const float*)d_in[0];
    const int*   ei       = (const int*)d_in[1];
    const float* W1       = (const float*)d_in[5];
    const float* att_src1 = (const float*)d_in[6];
    const float* att_dst1 = (const float*)d_in[7];
    const float* bias1    = (const float*)d_in[8];
    const float* W_lin    = (const float*)d_in[9];
    const float* b_lin    = (const float*)d_in[10];
    const float* W_att    = (const float*)d_in[11];
    const float* b_att    = (const float*)d_in[12];
    const float* W_le1    = (const float*)d_in[13];
    const float* b_le1    = (const float*)d_in[14];
    const float* W_le2    = (const float*)d_in[15];
    const float* W_le3    = (const float*)d_in[16];
    const float* b_le3    = (const float*)d_in[17];
    const float* W2       = (const float*)d_in[18];
    const float* att_src2 = (const float*)d_in[19];
    const float* att_dst2 = (const float*)d_in[20];
    const float* bias2    = (const float*)d_in[21];
    float* out = (float*)d_out;

    const long long NC = (long long)NND * CCH;             // 4,194,304
    float* ws = (float*)d_ws;
    size_t off = 0;
    auto alloc = [&](long long nflt) { size_t o = off; off += (size_t)((nflt + 63) & ~63LL); return o; };

    // big region: h|x1|xmax|xq, later overlaid by Ssel (both exactly 4*NC floats)
    size_t o_big   = alloc(4 * NC);
    float* h    = ws + o_big;
    float* x1   = ws + o_big + NC;
    float* xmax = ws + o_big + 2 * NC;
    float* xq   = ws + o_big + 3 * NC;
    float* Ssel = ws + o_big;                               // overlay after x1 dead
    float* xc   = ws + alloc(NC);
    float* Mbuf = ws + alloc((long long)NB * KSEL * NPG);   // 16.7M
    float* Acb  = ws + alloc((long long)NB * KSEL * KSEL);  // 8.4M
    float* Xsel = ws + alloc((long long)NB * KSEL * CCH);
    float* h2   = ws + alloc((long long)NB * KSEL * CCH);
    float* as_  = ws + alloc(NND * NHEAD);
    float* ad_  = ws + alloc(NND * NHEAD);
    float* mx1  = ws + alloc(NND * NHEAD);
    float* den1 = ws + alloc(NND * NHEAD);
    float* ex1  = ws + alloc((long long)ETC * NHEAD);
    float* q1   = ws + alloc(NND);
    float* q2   = ws + alloc(NND);
    float* mx2  = ws + alloc(NND);
    float* den2 = ws + alloc(NND);
    float* sex  = ws + alloc(ETC);
    float* a_n  = ws + alloc(NND);
    float* bb_n = ws + alloc(NND);
    float* agg  = ws + alloc(NND);
    float* cle3 = ws + alloc(NND);
    float* fit  = ws + alloc(NND);
    float* ds2  = ws + alloc(NB * KSEL);
    float* ss2  = ws + alloc(NB * KSEL);
    float* cm   = ws + alloc(NB * KSEL);
    int* selcol = (int*)(ws + alloc(NND));
    int* topi   = (int*)(ws + alloc(NB * KSEL));

    const unsigned int NEGINF = 0xFF800000u;
    const int EG = (ETC + 255) / 256;          // edge-parallel grid
    auto fill = [&](void* p, unsigned int bits, long long n) {
        fill_u32_kernel<<<1024, 256, 0, stream>>>((unsigned int*)p, bits, n);
    };

    // ---- init ----
    fill(mx1, NEGINF, NND * NHEAD);
    fill(den1, 0u, NND * NHEAD);
    fill(x1, 0u, NC);
    fill(xmax, NEGINF, NC);
    fill(mx2, NEGINF, NND);
    fill(den2, 0u, NND);
    fill(xc, 0u, NC);
    fill(agg, 0u, NND);
    fill(Mbuf, 0u, (long long)NB * KSEL * NPG);
    fill(cm, 0u, NB * KSEL);

    // ---- GAT1 ----
    launch_gemm(x, W1, h, NND, CCH, CCH, CCH, CCH, CCH, 0, 0, 0, 1, stream);
    node_head_dots<<<(NND * NHEAD + 255) / 256, 256, 0, stream>>>(h, att_src1, att_dst1, as_, ad_);
    edge_max1<<<EG, 256, 0, stream>>>(ei, as_, ad_, mx1);
    edge_exp1<<<EG, 256, 0, stream>>>(ei, as_, ad_, mx1, ex1, den1);
    x1_accum<<<ETC, CCH, 0, stream>>>(ei, h, ex1, den1, x1);
    relu_bias<<<(int)((NC + 255) / 256), 256, 0, stream>>>(x1, bias1);

    // ---- ASAPooling ----
    xmax_accum<<<ETC, CCH, 0, stream>>>(ei, x1, xmax);
    launch_gemm(xmax, W_lin, xq, NND, CCH, CCH, CCH, CCH, CCH, 0, 0, 0, 1, stream);
    q1q2_kernel<<<(NND + 255) / 256, 256, 0, stream>>>(xq, x1, b_lin, W_att, q1, q2);
    edge_max2<<<EG, 256, 0, stream>>>(ei, q1, q2, b_att, mx2);
    edge_exp2<<<EG, 256, 0, stream>>>(ei, q1, q2, b_att, mx2, sex, den2);
    score_scale<<<EG, 256, 0, stream>>>(ei, den2, sex);     // sex -> score
    xc_accum<<<ETC, CCH, 0, stream>>>(ei, x1, sex, xc);
    le_dots<<<(NND + 255) / 256, 256, 0, stream>>>(xc, W_le1, b_le1, W_le2, W_le3, b_le3,
                                                   a_n, bb_n, cle3);
    agg_accum<<<EG, 256, 0, stream>>>(ei, a_n, bb_n, agg);
    fitness_kernel<<<(NND + 255) / 256, 256, 0, stream>>>(agg, cle3, fit);
    topk_kernel<<<NB, 256, 0, stream>>>(fit, topi, selcol);
    xsel_build<<<(int)(((long long)NB * KSEL * CCH + 255) / 256), 256, 0, stream>>>(xc, fit, topi, Xsel);

    // ---- graph coarsening: Ac = (Ssel^T A) Ssel ----
    fill(Ssel, 0u, 4 * NC);                                  // overlay zero (h..xq dead)
    ssel_scatter<<<EG, 256, 0, stream>>>(ei, sex, selcol, Ssel);
    m_scatter<<<ETC, KSEL, 0, stream>>>(ei, Ssel, Mbuf);
    launch_gemm(Mbuf, Ssel, Acb, KSEL, KSEL, NPG, NPG, KSEL, KSEL,
                (long long)KSEL * NPG, (long long)NPG * KSEL, (long long)KSEL * KSEL,
                NB, stream);

    // ---- GAT2 (dense) + mean pool ----
    launch_gemm(Xsel, W2, h2, NB * KSEL, CCH, CCH, CCH, CCH, CCH, 0, 0, 0, 1, stream);
    h2_dots<<<(NB * KSEL + 255) / 256, 256, 0, stream>>>(h2, att_dst2, att_src2, ds2, ss2);
    gat2_softmax_cm<<<NB * KSEL, 128, 0, stream>>>(Acb, ds2, ss2, cm);
    final_out<<<NB, CCH, 0, stream>>>(cm, h2, bias2, out);

    (void)in_sizes; (void)n_in; (void)out_size; (void)ws_size;
}